// DilatedSelfAttention_20710332301568
// MI455X (gfx1250) — compile-verified
//
#include <hip/hip_runtime.h>
#include <math.h>

// ---------------------------------------------------------------------------
// Dilated self-attention for MI455X (gfx1250), fp32 WMMA (V_WMMA_F32_16X16X4_F32)
// b=4, n=8192, c=64; 7 windows/batch of m=2048; causal attention per window;
// denominator-weighted gather mix (scatter inverted analytically, no atomics).
// Round 3: all WMMA B operands are contiguous 8B LDS words (V / weights stored
// transposed, stride-68 padded) loaded as typed 64-bit (ds_load_b64, no
// component-marshaling movs); operand batching depth 8 to avoid spills.
// ---------------------------------------------------------------------------

typedef float v2f __attribute__((ext_vector_type(2)));
typedef float v8f __attribute__((ext_vector_type(8)));

#define WMMA_F32(A, B, Cc) \
  __builtin_amdgcn_wmma_f32_16x16x4_f32(false, (A), false, (B), (short)0, (Cc), false, false)

constexpr int Bb = 4;      // batch
constexpr int Nn = 8192;   // sequence length
constexpr int Cc = 64;     // channels
constexpr int Mm = 2048;   // window length
constexpr int NW = 7;      // windows per batch
constexpr int LP = 68;     // padded LDS row stride (floats): conflict-free

// Gather index for window w, element j (HEAD_IDX = 0)
__device__ __forceinline__ int widx(int w, int j) {
  if (w < 4) return w * 2048 + j;           // ws=2048, r=1
  if (w < 6) return (w - 4) * 4096 + 2 * j; // ws=4096, r=2
  return 4 * j;                             // ws=8192, r=4
}

__device__ __forceinline__ v8f v8zero() {
  v8f z = {0.f, 0.f, 0.f, 0.f, 0.f, 0.f, 0.f, 0.f};
  return z;
}

// Typed 64-bit LDS load: guarantees a single ds_load_b64 into an adjacent pair.
__device__ __forceinline__ v2f lds_ld64(const float* p) {
  union {
    double d;
    v2f v;
  } u;
  u.d = *(const double*)__builtin_assume_aligned(p, 8);
  return u.v;
}

// ---------------------------------------------------------------------------
// Kernel 1: gathered QKV projection. 128 thr = 4 waves; each wave = 16 rows.
// Weights staged TRANSPOSED in LDS: Wt[n*LP + k] = W[k*64 + n].
// Q is pre-scaled by 1/sqrt(c) = 0.125.
// ---------------------------------------------------------------------------
__global__ void qkv_kernel(const float* __restrict__ x,
                           const float* __restrict__ Wq,
                           const float* __restrict__ Wk,
                           const float* __restrict__ Wv,
                           float* __restrict__ Q, float* __restrict__ K,
                           float* __restrict__ V) {
  __shared__ float Wt[3][64 * LP];

  const int wave = threadIdx.x >> 5;
  const int lane = threadIdx.x & 31;
  const int tile = blockIdx.x * 4 + wave;           // global 16-row tile id
  const int rb   = tile % (Mm / 16);
  const int win  = (tile / (Mm / 16)) % NW;
  const int bb   = tile / ((Mm / 16) * NW);
  const int r0   = rb * 16;
  const int l15  = lane & 15;
  const int half = lane >> 4;
  const int koff = half * 2;

  // cooperative transpose-copy of the three 64x64 weight matrices
  {
    const float4* srcs[3] = {(const float4*)Wq, (const float4*)Wk,
                             (const float4*)Wv};
#pragma unroll
    for (int mtx = 0; mtx < 3; ++mtx) {
      const float4* s = srcs[mtx];
      for (int i = threadIdx.x; i < 1024; i += 128) {
        const int row = i >> 4, c4 = (i & 15) << 2; // W[row][c4..c4+3]
        const float4 w = s[i];
        Wt[mtx][(c4 + 0) * LP + row] = w.x;
        Wt[mtx][(c4 + 1) * LP + row] = w.y;
        Wt[mtx][(c4 + 2) * LP + row] = w.z;
        Wt[mtx][(c4 + 3) * LP + row] = w.w;
      }
    }
  }

  // A operand: gathered x rows in WMMA f32 A layout
  const int g = widx(win, r0 + l15);
  const float* xrow = x + ((size_t)bb * Nn + g) * Cc;
  v2f a[16];
#pragma unroll
  for (int t = 0; t < 16; ++t) {
    a[t].x = xrow[4 * t + koff];
    a[t].y = xrow[4 * t + koff + 1];
  }
  __syncthreads();

  float* Outp[3] = {Q, K, V};
  const size_t obase = (((size_t)bb * NW + win) * Mm + r0) * Cc;

  for (int mtx = 0; mtx < 3; ++mtx) {
    const float* W = Wt[mtx];
    const float sc = (mtx == 0) ? 0.125f : 1.0f;
#pragma unroll
    for (int nt = 0; nt < 4; ++nt) {
      const int n = nt * 16 + l15;
      v8f acc = v8zero();
#pragma unroll
      for (int ch = 0; ch < 2; ++ch) { // batch 8 B operands per chunk
        v2f b[8];
#pragma unroll
        for (int j = 0; j < 8; ++j)
          b[j] = lds_ld64(&W[n * LP + 4 * (8 * ch + j) + koff]);
#pragma unroll
        for (int j = 0; j < 8; ++j) acc = WMMA_F32(a[8 * ch + j], b[j], acc);
      }
      float* o = Outp[mtx] + obase;
#pragma unroll
      for (int v = 0; v < 8; ++v)
        o[(size_t)(v + 8 * half) * Cc + n] = acc[v] * sc;
    }
  }
}

// ---------------------------------------------------------------------------
// Kernel 2: flash attention per window. 128 thr = 4 waves; wave = 16 q rows;
// workgroup = 64 q rows; loop over 64-key blocks (causal).
// Kb row-major (B reads contiguous); V stored TRANSPOSED (Vt[n][k]).
// ---------------------------------------------------------------------------
__global__ void attn_kernel(const float* __restrict__ Q,
                            const float* __restrict__ K,
                            const float* __restrict__ V,
                            float* __restrict__ O, float* __restrict__ Dn) {
  __shared__ float Kb[64 * LP];
  __shared__ float Vt[64 * LP];
  __shared__ float Pst[4 * 16 * LP];

  const int wave = threadIdx.x >> 5;
  const int lane = threadIdx.x & 31;
  const int l15  = lane & 15;
  const int half = lane >> 4;
  const int koff = half * 2;

  const int qb = blockIdx.x & 31;        // q block (64 rows), 32 per window
  const int wg = blockIdx.x >> 5;        // window id = bb*NW + win
  const size_t base = (size_t)wg * Mm * Cc;
  const int qrow0 = qb * 64 + wave * 16; // wave's first q row

  // Q in A layout (pre-scaled in kernel 1)
  v2f qa[16];
  {
    const float* qp = Q + base + (size_t)(qrow0 + l15) * Cc;
#pragma unroll
    for (int t = 0; t < 16; ++t) {
      qa[t].x = qp[4 * t + koff];
      qa[t].y = qp[4 * t + koff + 1];
    }
  }

  float mrow[8], srow[8];
  v8f o_acc[4];
#pragma unroll
  for (int v = 0; v < 8; ++v) { mrow[v] = -__builtin_inff(); srow[v] = 0.f; }
#pragma unroll
  for (int nt = 0; nt < 4; ++nt) o_acc[nt] = v8zero();

  float* Pw = Pst + wave * 16 * LP;

  for (int kb = 0; kb <= qb; ++kb) {
    __syncthreads();
    { // cooperative K (row-major) / V (transposed) block load into padded LDS
      const float4* ks = (const float4*)(K + base + (size_t)kb * 64 * Cc);
      const float4* vs = (const float4*)(V + base + (size_t)kb * 64 * Cc);
      for (int i = threadIdx.x; i < 1024; i += 128) {
        const int row = i >> 4, c4 = (i & 15) << 2;
        *(float4*)&Kb[row * LP + c4] = ks[i];
        const float4 v = vs[i];
        Vt[(c4 + 0) * LP + row] = v.x;
        Vt[(c4 + 1) * LP + row] = v.y;
        Vt[(c4 + 2) * LP + row] = v.z;
        Vt[(c4 + 3) * LP + row] = v.w;
      }
      // warm L2/WGP$ for next block (global_prefetch_b8)
      if (kb < qb) {
        const char* pk = (const char*)(K + base + (size_t)(kb + 1) * 64 * Cc) +
                         (size_t)threadIdx.x * 128;
        const char* pv = (const char*)(V + base + (size_t)(kb + 1) * 64 * Cc) +
                         (size_t)threadIdx.x * 128;
        __builtin_prefetch(pk, 0, 1);
        __builtin_prefetch(pv, 0, 1);
      }
    }
    __syncthreads();

    // logits: S[16q][64k] as 4 WMMA C tiles; B operands = contiguous K rows
    v8f st[4];
#pragma unroll
    for (int kt = 0; kt < 4; ++kt) {
      const int krow = kt * 16 + l15;
      v8f acc = v8zero();
#pragma unroll
      for (int ch = 0; ch < 2; ++ch) {
        v2f b[8];
#pragma unroll
        for (int j = 0; j < 8; ++j)
          b[j] = lds_ld64(&Kb[krow * LP + 4 * (8 * ch + j) + koff]);
#pragma unroll
        for (int j = 0; j < 8; ++j) acc = WMMA_F32(qa[8 * ch + j], b[j], acc);
      }
      st[kt] = acc;
    }

    if (kb == qb) { // causal mask on the diagonal block
#pragma unroll
      for (int kt = 0; kt < 4; ++kt) {
        const int key = kb * 64 + kt * 16 + l15;
#pragma unroll
        for (int v = 0; v < 8; ++v)
          if (key > qrow0 + v + 8 * half) st[kt][v] = -__builtin_inff();
      }
    }

    // online softmax (row = vgpr v + 8*half; N values live in 16-lane half)
    float fscale[8], ladd[8];
#pragma unroll
    for (int v = 0; v < 8; ++v) {
      float mx = fmaxf(fmaxf(st[0][v], st[1][v]), fmaxf(st[2][v], st[3][v]));
      for (int off = 1; off < 16; off <<= 1)
        mx = fmaxf(mx, __shfl_xor(mx, off, 32));
      const float mn = fmaxf(mrow[v], mx);
      fscale[v] = __expf(mrow[v] - mn);
      mrow[v] = mn;
      ladd[v] = 0.f;
    }
#pragma unroll
    for (int kt = 0; kt < 4; ++kt)
#pragma unroll
      for (int v = 0; v < 8; ++v) {
        const float p = __expf(st[kt][v] - mrow[v]); // -inf -> 0
        st[kt][v] = p;
        ladd[v] += p;
      }
#pragma unroll
    for (int v = 0; v < 8; ++v) {
      float s = ladd[v];
      for (int off = 1; off < 16; off <<= 1) s += __shfl_xor(s, off, 32);
      srow[v] = srow[v] * fscale[v] + s;
#pragma unroll
      for (int nt = 0; nt < 4; ++nt) o_acc[nt][v] *= fscale[v];
    }

    // C-layout -> A-layout transpose of P through padded LDS
#pragma unroll
    for (int kt = 0; kt < 4; ++kt) {
      const int col = kt * 16 + l15;
#pragma unroll
      for (int v = 0; v < 8; ++v) Pw[(v + 8 * half) * LP + col] = st[kt][v];
    }
    __syncthreads();

    // O += P x V; A = P rows (contiguous), B = Vt rows (contiguous)
    v2f pa[16];
#pragma unroll
    for (int t = 0; t < 16; ++t)
      pa[t] = lds_ld64(&Pw[l15 * LP + 4 * t + koff]);
#pragma unroll
    for (int nt = 0; nt < 4; ++nt) {
      const int nc = nt * 16 + l15;
      v8f acc = o_acc[nt];
#pragma unroll
      for (int ch = 0; ch < 2; ++ch) {
        v2f b[8];
#pragma unroll
        for (int j = 0; j < 8; ++j)
          b[j] = lds_ld64(&Vt[nc * LP + 4 * (8 * ch + j) + koff]);
#pragma unroll
        for (int j = 0; j < 8; ++j) acc = WMMA_F32(pa[8 * ch + j], b[j], acc);
      }
      o_acc[nt] = acc;
    }
  }

  // normalize + store o and denominators
  float inv[8];
#pragma unroll
  for (int v = 0; v < 8; ++v) inv[v] = 1.f / srow[v];
  const size_t orow = base + (size_t)qrow0 * Cc;
#pragma unroll
  for (int nt = 0; nt < 4; ++nt) {
    const int nc = nt * 16 + l15;
#pragma unroll
    for (int v = 0; v < 8; ++v)
      O[orow + (size_t)(v + 8 * half) * Cc + nc] = o_acc[nt][v] * inv[v];
  }
  if (l15 == 0) {
#pragma unroll
    for (int v = 0; v < 8; ++v)
      Dn[(size_t)wg * Mm + qrow0 + v + 8 * half] = srow[v] * __expf(mrow[v]);
  }
}

// ---------------------------------------------------------------------------
// Kernel 3: gather mix. out[pos] = sum_i d_i * o_i / sum_i d_i.
// Contributors of pos: w=pos>>11 always; w=4+(pos>>12) if pos even; w=6 if %4.
// ---------------------------------------------------------------------------
__device__ __forceinline__ void accum_contrib(const float* __restrict__ Ow,
                                              const float* __restrict__ Dn,
                                              int bb, int w, int j, int c0,
                                              float* acc, float& ds) {
  const size_t ri = ((size_t)bb * NW + w) * (size_t)Mm + j;
  const float d = Dn[ri];
  ds += d;
  const float4* o4 = (const float4*)(Ow + ri * (size_t)Cc + c0);
#pragma unroll
  for (int i = 0; i < 4; ++i) {
    const float4 t = o4[i];
    acc[4 * i + 0] += d * t.x;
    acc[4 * i + 1] += d * t.y;
    acc[4 * i + 2] += d * t.z;
    acc[4 * i + 3] += d * t.w;
  }
}

__global__ void mix_kernel(const float* __restrict__ Ow,
                           const float* __restrict__ Dn,
                           float* __restrict__ out) {
  const int gid = blockIdx.x * 256 + threadIdx.x;
  if (gid >= Bb * Nn * 4) return;
  const int c0 = (gid & 3) * 16;
  const int pn = gid >> 2;
  const int pos = pn & (Nn - 1);
  const int bb = pn >> 13; // Nn = 2^13

  float acc[16];
#pragma unroll
  for (int i = 0; i < 16; ++i) acc[i] = 0.f;
  float ds = 0.f;

  accum_contrib(Ow, Dn, bb, pos >> 11, pos & 2047, c0, acc, ds);
  if (!(pos & 1))
    accum_contrib(Ow, Dn, bb, 4 + (pos >> 12), (pos & 4095) >> 1, c0, acc, ds);
  if (!(pos & 3)) accum_contrib(Ow, Dn, bb, 6, pos >> 2, c0, acc, ds);

  const float invds = 1.f / ds;
  float4* dst = (float4*)(out + ((size_t)bb * Nn + pos) * Cc + c0);
#pragma unroll
  for (int i = 0; i < 4; ++i) {
    float4 t;
    t.x = acc[4 * i + 0] * invds;
    t.y = acc[4 * i + 1] * invds;
    t.z = acc[4 * i + 2] * invds;
    t.w = acc[4 * i + 3] * invds;
    dst[i] = t;
  }
}

// ---------------------------------------------------------------------------
extern "C" void kernel_launch(void* const* d_in, const int* in_sizes, int n_in,
                              void* d_out, int out_size, void* d_ws,
                              size_t ws_size, hipStream_t stream) {
  const float* x  = (const float*)d_in[0];
  const float* Wq = (const float*)d_in[1];
  const float* Wk = (const float*)d_in[2];
  const float* Wv = (const float*)d_in[3];
  float* out = (float*)d_out;

  // workspace layout (floats): Q | K | V | O | Dn  (~59 MB total)
  const size_t szw = (size_t)Bb * NW * Mm * Cc; // 3,670,016
  float* Q  = (float*)d_ws;
  float* K  = Q + szw;
  float* V  = K + szw;
  float* Ow = V + szw;
  float* Dn = Ow + szw;

  // 1) gathered QKV projection: 3584 wave-tiles, 4 per 128-thread block
  qkv_kernel<<<(Bb * NW * (Mm / 16)) / 4, 128, 0, stream>>>(x, Wq, Wk, Wv, Q,
                                                            K, V);

  // 2) flash attention: 28 windows x 32 q-blocks
  attn_kernel<<<Bb * NW * 32, 128, 0, stream>>>(Q, K, V, Ow, Dn);

  // 3) gather mix
  mix_kernel<<<(Bb * Nn * 4 + 255) / 256, 256, 0, stream>>>(Ow, Dn, out);
}